// HGV_66108136620050
// MI455X (gfx1250) — compile-verified
//
#include <hip/hip_runtime.h>
#include <hip/hip_bf16.h>
#include <math.h>

// Problem constants (reference: B,T,D,H = 2048,48,1024,1024)
#define BB 2048
#define TT 48
#define DD 1024
#define HH 1024

typedef __attribute__((ext_vector_type(2))) float v2f;
typedef __attribute__((ext_vector_type(8))) float v8f;
typedef unsigned int u32;
typedef __attribute__((ext_vector_type(4))) u32 u32x4;
typedef __attribute__((ext_vector_type(4))) int i32x4;
typedef __attribute__((ext_vector_type(8))) int i32x8;

#define HAVE_WMMA_F32X4 (__has_builtin(__builtin_amdgcn_wmma_f32_16x16x4_f32))
#define HAVE_TDM (__has_builtin(__builtin_amdgcn_tensor_load_to_lds))
#if __has_include(<hip/amd_detail/amd_gfx1250_TDM.h>)
#define TDM_6ARG 1   // therock-10.0 headers -> clang-23 6-arg builtin
#else
#define TDM_6ARG 0   // ROCm 7.2 clang-22 5-arg builtin
#endif

// ---------------------------------------------------------------------------
// Kernel 1: M = Wt @ Wx^T   [D x D],  Wt,Wx are [D,H] row-major.
// One wave32 per 16x16 output tile. grid = (D/16, D/16), block = 32.
// WMMA f32 layouts (ISA 7.12.2):
//   A 16x4 : lane=16*half+lm, reg r -> A[M=lm, K=2*half+r]
//   B 4x16 : reg r -> B[K=2*half+r, N=lm]
//   C 16x16: reg r -> C[M=r+8*half, N=lm]
// ---------------------------------------------------------------------------
__global__ __launch_bounds__(32) void k1_wtwxT(const float* __restrict__ Wt,
                                               const float* __restrict__ Wx,
                                               float* __restrict__ Mo) {
    const int lane = threadIdx.x;
    const int lm   = lane & 15;
    const int half = lane >> 4;
    const int i0 = blockIdx.x * 16;   // rows of M (from Wt)
    const int j0 = blockIdx.y * 16;   // cols of M (from Wx)
    v8f acc = {};
#if HAVE_WMMA_F32X4
    const float* arow = Wt + (size_t)(i0 + lm) * HH;  // A[M=lm, k...]
    const float* brow = Wx + (size_t)(j0 + lm) * HH;  // B[K,N=lm] = Wx[j0+lm, k+K]
    for (int k = 0; k < HH; k += 4) {
        v2f A = *(const v2f*)(arow + k + 2 * half);   // K = 2*half + {0,1}
        v2f Bv = *(const v2f*)(brow + k + 2 * half);
        acc = __builtin_amdgcn_wmma_f32_16x16x4_f32(
            false, A, false, Bv, (short)0, acc, false, false);
    }
#else
    for (int k = 0; k < HH; ++k) {
        const float bv = Wx[(size_t)(j0 + lm) * HH + k];
#pragma unroll
        for (int r = 0; r < 8; ++r)
            acc[r] = fmaf(Wt[(size_t)(i0 + r + 8 * half) * HH + k], bv, acc[r]);
    }
#endif
#pragma unroll
    for (int r = 0; r < 8; ++r)
        Mo[(size_t)(i0 + r + 8 * half) * DD + j0 + lm] = acc[r];
}

// ---------------------------------------------------------------------------
// Kernel 2: P = X_last @ M   [B x D],  X_last[b,:] = input[b, T-1, :].
// One wave32 per 16x16 tile. grid = (B/16, D/16), block = 32.
// ---------------------------------------------------------------------------
__global__ __launch_bounds__(32) void k2_xlastM(const float* __restrict__ input,
                                                const float* __restrict__ Mo,
                                                float* __restrict__ P) {
    const int lane = threadIdx.x;
    const int lm   = lane & 15;
    const int half = lane >> 4;
    const int i0 = blockIdx.x * 16;   // batch rows
    const int j0 = blockIdx.y * 16;   // output cols
    v8f acc = {};
#if HAVE_WMMA_F32X4
    const float* arow = input + (size_t)(i0 + lm) * (TT * DD) + (TT - 1) * DD;
    for (int k = 0; k < DD; k += 4) {
        v2f A = *(const v2f*)(arow + k + 2 * half);
        v2f Bv;
        Bv.x = Mo[(size_t)(k + 2 * half + 0) * DD + j0 + lm];
        Bv.y = Mo[(size_t)(k + 2 * half + 1) * DD + j0 + lm];
        acc = __builtin_amdgcn_wmma_f32_16x16x4_f32(
            false, A, false, Bv, (short)0, acc, false, false);
    }
#else
    for (int k = 0; k < DD; ++k) {
        const float bv = Mo[(size_t)k * DD + j0 + lm];
#pragma unroll
        for (int r = 0; r < 8; ++r)
            acc[r] = fmaf(input[(size_t)(i0 + r + 8 * half) * (TT * DD) + (TT - 1) * DD + k],
                          bv, acc[r]);
    }
#endif
#pragma unroll
    for (int r = 0; r < 8; ++r)
        P[(size_t)(i0 + r + 8 * half) * DD + j0 + lm] = acc[r];
}

// ---------------------------------------------------------------------------
// Kernel 3 (bandwidth-dominant): one workgroup per batch row b.
// Stages input[b] (192 KB) into LDS via the Tensor Data Mover (one
// tensor_load_to_lds per block, TENSORcnt-tracked) -> dot -> scalar chain
// -> softmax -> v.  Reads `input` exactly once from HBM (~402 MB total):
// ~18 us roofline at 23.3 TB/s.
// 512 threads = 16 wave32s; LDS use ~201 KB of the 320 KB/WGP pool.
// ---------------------------------------------------------------------------
__global__ __launch_bounds__(512) void k3_fused(const float* __restrict__ input,
                                                const float* __restrict__ input2,
                                                const float* __restrict__ P,
                                                const float* __restrict__ rate,
                                                const float* __restrict__ rate2,
                                                float* __restrict__ out_v,
                                                float* __restrict__ out_a) {
    extern __shared__ float smem[];
    float* xs   = smem;                 // [TT*DD]  = 49152 floats (192 KB)
    float* pv   = xs + TT * DD;         // [DD]     = 1024
    float* dotv = pv + DD;              // [TT]
    float* av   = dotv + TT;            // [TT]

    const int b    = blockIdx.x;
    const int tid  = threadIdx.x;
    const int lane = tid & 31;
    const int wave = tid >> 5;          // 0..15

#if HAVE_TDM
    // --- TDM: one DMA descriptor moves input[b] (48*1024 f32) to LDS. ---
    if (wave == 0) {
        const unsigned long long ga =
            (unsigned long long)(const void*)(input + (size_t)b * TT * DD);
        const u32 lds_off = (u32)(unsigned long long)(void*)xs; // generic->LDS byte offset
        // D# group 0 (ISA 08 §8.3): [1:0]=count=1, [63:32]=lds_addr,
        // [120:64]=global_addr, [127:126]=type=2.
        u32x4 g0;
        g0.x = 1u;
        g0.y = lds_off;
        g0.z = (u32)ga;
        g0.w = (u32)((ga >> 32) & 0x01FFFFFFu) | (2u << 30);
        // D# group 1 (§8.4): data_size=2 (4B); tensor_dim0=tile_dim0=49152
        // (one row of 192 KB); tensor_dim1=tile_dim1=1; dim0_stride=49152.
        const u32 n0 = (u32)(TT * DD);            // 0xC000
        i32x8 g1;
        g1[0] = (int)(2u << 16);                  // workgroup_mask=0, data_size=4B
        g1[1] = (int)((n0 & 0xFFFFu) << 16);      // tensor_dim0[15:0] @ bits 63:48
        g1[2] = (int)((n0 >> 16) | (1u << 16));   // tensor_dim0[31:16] | tensor_dim1[15:0]
        g1[3] = (int)(n0 << 16);                  // tensor_dim1[31:16]=0 | tile_dim0 @127:112
        g1[4] = 1;                                // tile_dim1=1, tile_dim2=0
        g1[5] = (int)n0;                          // tensor_dim0_stride[31:0]
        g1[6] = 0;                                // stride hi / dim1_stride lo
        g1[7] = 0;
        const i32x4 z4 = {};
#if TDM_6ARG
        const i32x8 z8 = {};
        __builtin_amdgcn_tensor_load_to_lds(g0, g1, z4, z4, z8, 0);
#else
        __builtin_amdgcn_tensor_load_to_lds(g0, g1, z4, z4, 0);
#endif
    }
    // Overlap: all threads stage P[b,:] (4 KB) while the TDM streams.
    ((float2*)pv)[tid] = ((const float2*)(P + (size_t)b * DD))[tid];
    if (wave == 0) __builtin_amdgcn_s_wait_tensorcnt(0);
    __syncthreads();
#else
    // Fallback: coalesced b128 copy through VGPRs.
    ((float2*)pv)[tid] = ((const float2*)(P + (size_t)b * DD))[tid];
    {
        const float4* src = (const float4*)(input + (size_t)b * TT * DD);
        float4* dst = (float4*)xs;
#pragma unroll
        for (int i = 0; i < (TT * DD / 4) / 512; ++i) {
            const int idx = tid + 512 * i;
            dst[idx] = src[idx];
        }
    }
    __syncthreads();
#endif

    // dot[t] = P[b,:] . x[b,t,:]  — wave w handles t = w, w+16, w+32.
    for (int t = wave; t < TT; t += 16) {
        const float* xr = xs + (size_t)t * DD;
        float s = 0.f;
#pragma unroll
        for (int i = 0; i < DD / 32; ++i) {
            const int d = lane + 32 * i;
            s = fmaf(xr[d], pv[d], s);
        }
#pragma unroll
        for (int off = 16; off; off >>= 1) s += __shfl_xor(s, off, 32);
        if (lane == 0) dotv[t] = s;
    }
    __syncthreads();

    // Wave 0: scalar chain + softmax over the 48 time steps.
    if (wave == 0) {
        const float r   = rate[0];
        const float sr2 = 1.f / (1.f + __expf(-rate2[0]));
        const float* in2 = input2 + (size_t)b * TT;

        // max_t |input2[b,t]|
        float m = fabsf(in2[lane]);
        if (lane < 16) m = fmaxf(m, fabsf(in2[32 + lane]));
#pragma unroll
        for (int off = 16; off; off >>= 1) m = fmaxf(m, __shfl_xor(m, off, 32));
        const float sig_max = 1.f / (1.f + __expf(-m));

        auto calc_e = [&](int t) -> float {
            const float sig = 1.f / (1.f + __expf(-dotv[t]));
            const float btd = (float)(TT - 1 - t) / (float)TT + 0.01f;
            const float bvd = (1.f / (1.f + __expf(-in2[t]))) / sig_max;
            const float fb  = (1.f + sr2) * btd * bvd / (sr2 * btd + bvd);
            const float den = r * (__logf(2.72f + (1.f - sig)) * fb) * (float)TT;
            return fmaxf(0.f, sig / den);
        };

        const bool has2 = lane < 16;
        const float e0 = calc_e(lane);
        const float e1 = has2 ? calc_e(32 + lane) : 0.f;

        float mx = has2 ? fmaxf(e0, e1) : e0;
#pragma unroll
        for (int off = 16; off; off >>= 1) mx = fmaxf(mx, __shfl_xor(mx, off, 32));
        const float x0 = __expf(e0 - mx);
        const float x1 = has2 ? __expf(e1 - mx) : 0.f;
        float ssum = x0 + x1;
#pragma unroll
        for (int off = 16; off; off >>= 1) ssum += __shfl_xor(ssum, off, 32);
        const float inv = 1.f / ssum;

        av[lane] = x0 * inv;
        out_a[(size_t)b * TT + lane] = x0 * inv;
        if (has2) {
            av[32 + lane] = x1 * inv;
            out_a[(size_t)b * TT + 32 + lane] = x1 * inv;
        }
    }
    __syncthreads();

    // v[b,d] = sum_t a[t] * x[b,t,d] — each thread covers d = tid, tid+512.
    float acc0 = 0.f, acc1 = 0.f;
    const int d0 = tid, d1 = tid + 512;
#pragma unroll
    for (int t = 0; t < TT; ++t) {
        const float a = av[t];
        acc0 = fmaf(a, xs[(size_t)t * DD + d0], acc0);
        acc1 = fmaf(a, xs[(size_t)t * DD + d1], acc1);
    }
    out_v[(size_t)b * DD + d0] = acc0;
    out_v[(size_t)b * DD + d1] = acc1;
}

// ---------------------------------------------------------------------------
extern "C" void kernel_launch(void* const* d_in, const int* in_sizes, int n_in,
                              void* d_out, int out_size, void* d_ws, size_t ws_size,
                              hipStream_t stream) {
    (void)in_sizes; (void)n_in; (void)out_size; (void)ws_size;
    const float* input  = (const float*)d_in[0];   // [B,T,D]
    const float* input2 = (const float*)d_in[1];   // [B,T]
    const float* Wt     = (const float*)d_in[2];   // [D,H]
    const float* Wx     = (const float*)d_in[3];   // [D,H]
    const float* rate   = (const float*)d_in[4];   // [1]
    const float* rate2  = (const float*)d_in[5];   // [1]
    // d_in[6], d_in[7] (each_epoch, step) unused by the reference math.

    float* Mo = (float*)d_ws;                        // [D,D]  4 MB
    float* P  = (float*)d_ws + (size_t)DD * DD;      // [B,D]  8 MB
    float* out_v = (float*)d_out;                    // [B,D]
    float* out_a = out_v + (size_t)BB * DD;          // [B,T]

    k1_wtwxT<<<dim3(DD / 16, DD / 16), 32, 0, stream>>>(Wt, Wx, Mo);
    k2_xlastM<<<dim3(BB / 16, DD / 16), 32, 0, stream>>>(input, Mo, P);

    const size_t smem = (size_t)(TT * DD + DD + TT + TT) * sizeof(float); // ~201 KB
    k3_fused<<<BB, 512, smem, stream>>>(input, input2, P, rate, rate2, out_v, out_a);
}